// GATEncoder_52355651338866
// MI455X (gfx1250) — compile-verified
//
#include <hip/hip_runtime.h>
#include <hip/hip_bf16.h>

#define N_NODES   100000
#define N_EDGES   3200000
#define N_GRAPHS  2048
#define IN_DIM    128
#define HIDDEN    64
#define OUT_DIM   128
#define HEADS     4
#define E_TOT     (N_EDGES + N_NODES)
#define NEG_SLOPE 0.2f

typedef float v2f __attribute__((ext_vector_type(2)));
typedef float v8f __attribute__((ext_vector_type(8)));

// ---------- helpers: order-preserving float<->uint encoding for atomic max ----
__device__ __forceinline__ unsigned fenc(float f) {
    unsigned u = __float_as_uint(f);
    return (u & 0x80000000u) ? ~u : (u | 0x80000000u);
}
__device__ __forceinline__ float fdec(unsigned u) {
    return __uint_as_float((u & 0x80000000u) ? (u ^ 0x80000000u) : ~u);
}
__device__ __forceinline__ float lrelu(float v) {
    return v > 0.0f ? v : NEG_SLOPE * v;
}

// ---------------------------- fills ------------------------------------------
__global__ void fill_f32(float* __restrict__ p, float v, int n) {
    int i = blockIdx.x * blockDim.x + threadIdx.x;
    if (i < n) p[i] = v;
}
__global__ void fill_u32(unsigned* __restrict__ p, unsigned v, int n) {
    int i = blockIdx.x * blockDim.x + threadIdx.x;
    if (i < n) p[i] = v;
}

// ----------------- fp32 GEMM via V_WMMA_F32_16X16X4_F32 ----------------------
// C[M,N] = A[M,K] @ B[K,N], row-major, M%16==0, N==16*NT, K%4==0.
// One wave32 per 16-row strip: the A fragment is loaded ONCE per k-step and
// swept across all NT n-tiles, with NT accumulator tiles held in registers.
// A frag (16x4 f32): lane holds A[row][k + 2*(lane>>4) + j], j=0,1
// B frag (4x16 f32): lane holds B[k + 2*(lane>>4) + j][col]
// C/D (16x16 f32): VGPR i <- C[tm*16 + i + 8*(lane>>4)][tn*16 + (lane&15)]
template <int NT>
__global__ __launch_bounds__(32)
void gemm_wmma_f32_strip(const float* __restrict__ A, const float* __restrict__ B,
                         float* __restrict__ C, int K) {
    const int N    = NT * 16;
    const int lane = threadIdx.x;
    const int half = lane >> 4;
    const int l16  = lane & 15;
    const int tm   = blockIdx.x;
    const int arow = tm * 16 + l16;

    v8f acc[NT];
#pragma unroll
    for (int t = 0; t < NT; ++t) acc[t] = {};

    for (int k = 0; k < K; k += 4) {
        // A fragment: contiguous K-pair, 8B-aligned (K even, k%4==0)
        v2f a = *(const v2f*)(A + (size_t)arow * K + k + 2 * half);
        const float* bp = B + (size_t)(k + 2 * half) * N + l16;
#pragma unroll
        for (int t = 0; t < NT; ++t) {
            v2f b;
            b.x = bp[t * 16];
            b.y = bp[t * 16 + N];
            acc[t] = __builtin_amdgcn_wmma_f32_16x16x4_f32(
                /*neg_a=*/false, a, /*neg_b=*/false, b,
                /*c_mod=*/(short)0, acc[t],
                /*reuse_a=*/false, /*reuse_b=*/false);
        }
    }
#pragma unroll
    for (int t = 0; t < NT; ++t) {
        float* cp = C + (size_t)(tm * 16 + 8 * half) * N + t * 16 + l16;
#pragma unroll
        for (int i = 0; i < 8; ++i) cp[(size_t)i * N] = acc[t][i];
    }
}

// -------------------- per-node attention logits ------------------------------
// al_s[n,h] = sum_c h[n,h,c]*a_src[h,c] ; al_d likewise. One thread per (n,h).
__global__ void att_logits(const float* __restrict__ h,
                           const float* __restrict__ a_src,
                           const float* __restrict__ a_dst,
                           float* __restrict__ al_s, float* __restrict__ al_d,
                           int C) {
    int i = blockIdx.x * blockDim.x + threadIdx.x;
    if (i >= N_NODES * HEADS) return;
    int hd = i & (HEADS - 1);
    int n  = i >> 2;
    const float* hp = h + (size_t)n * (HEADS * C) + hd * C;
    const float* as = a_src + hd * C;
    const float* ad = a_dst + hd * C;
    float ss = 0.0f, sd = 0.0f;
    for (int c = 0; c < C; ++c) { float v = hp[c]; ss += v * as[c]; sd += v * ad[c]; }
    al_s[i] = ss;
    al_d[i] = sd;
}

// -------------------------- edge passes --------------------------------------
__device__ __forceinline__ void edge_sd(long long e, const int* __restrict__ srcs,
                                        const int* __restrict__ dsts, int& s, int& d) {
    if (e < N_EDGES) { s = srcs[e]; d = dsts[e]; }
    else             { s = d = (int)(e - N_EDGES); }  // self loop
}

__global__ void edge_max(const int* __restrict__ srcs, const int* __restrict__ dsts,
                         const float* __restrict__ al_s, const float* __restrict__ al_d,
                         unsigned* __restrict__ menc) {
    long long i = (long long)blockIdx.x * blockDim.x + threadIdx.x;
    if (i >= (long long)E_TOT * HEADS) return;
    int hd = (int)(i & (HEADS - 1));
    long long e = i >> 2;
    int s, d; edge_sd(e, srcs, dsts, s, d);
    float v = lrelu(al_s[s * HEADS + hd] + al_d[d * HEADS + hd]);
    atomicMax(&menc[d * HEADS + hd], fenc(v));
}

__global__ void edge_expsum(const int* __restrict__ srcs, const int* __restrict__ dsts,
                            const float* __restrict__ al_s, const float* __restrict__ al_d,
                            const unsigned* __restrict__ menc, float* __restrict__ den) {
    long long i = (long long)blockIdx.x * blockDim.x + threadIdx.x;
    if (i >= (long long)E_TOT * HEADS) return;
    int hd = (int)(i & (HEADS - 1));
    long long e = i >> 2;
    int s, d; edge_sd(e, srcs, dsts, s, d);
    float v = lrelu(al_s[s * HEADS + hd] + al_d[d * HEADS + hd]);
    float m = fdec(menc[d * HEADS + hd]);
    atomicAdd(&den[d * HEADS + hd], __expf(v - m));
}

__global__ void edge_agg(const int* __restrict__ srcs, const int* __restrict__ dsts,
                         const float* __restrict__ al_s, const float* __restrict__ al_d,
                         const unsigned* __restrict__ menc, const float* __restrict__ den,
                         const float* __restrict__ h, float* __restrict__ agg, int C) {
    long long i = (long long)blockIdx.x * blockDim.x + threadIdx.x;
    if (i >= (long long)E_TOT * HEADS) return;
    int hd = (int)(i & (HEADS - 1));
    long long e = i >> 2;
    int s, d; edge_sd(e, srcs, dsts, s, d);
    float v = lrelu(al_s[s * HEADS + hd] + al_d[d * HEADS + hd]);
    float m = fdec(menc[d * HEADS + hd]);
    float alpha = __expf(v - m) / (den[d * HEADS + hd] + 1e-16f);
    const float* hp = h   + (size_t)s * (HEADS * C) + hd * C;
    float*       op = agg + (size_t)d * (HEADS * C) + hd * C;
    __builtin_prefetch(hp, 0, 1);   // global_prefetch_b8
    for (int c = 0; c < C; ++c) atomicAdd(&op[c], alpha * hp[c]);
}

// -------------------------- bias + relu (in place) ---------------------------
__global__ void bias_relu(float* __restrict__ x, const float* __restrict__ b,
                          int Fmask, int n) {
    int i = blockIdx.x * blockDim.x + threadIdx.x;
    if (i >= n) return;
    float v = x[i] + b[i & Fmask];
    x[i] = v > 0.0f ? v : 0.0f;
}

// -------------------------- global mean pool ---------------------------------
__global__ void pool_acc(const float* __restrict__ h, const int* __restrict__ batch,
                         float* __restrict__ sums, float* __restrict__ cnt) {
    int i = blockIdx.x * blockDim.x + threadIdx.x;
    if (i >= N_NODES * OUT_DIM) return;
    int f = i & (OUT_DIM - 1);
    int n = i >> 7;
    int g = batch[n];
    atomicAdd(&sums[g * OUT_DIM + f], h[i]);
    if (f == 0) atomicAdd(&cnt[g], 1.0f);
}
__global__ void pool_div(const float* __restrict__ sums, const float* __restrict__ cnt,
                         float* __restrict__ out) {
    int i = blockIdx.x * blockDim.x + threadIdx.x;
    if (i >= N_GRAPHS * OUT_DIM) return;
    float c = cnt[i >> 7];
    out[i] = sums[i] / (c < 1.0f ? 1.0f : c);
}

// ============================== launcher =====================================
extern "C" void kernel_launch(void* const* d_in, const int* in_sizes, int n_in,
                              void* d_out, int out_size, void* d_ws, size_t ws_size,
                              hipStream_t stream) {
    const float* x     = (const float*)d_in[0];
    const int*   ei    = (const int*)  d_in[1];   // [2, E]
    const int*   batch = (const int*)  d_in[3];
    const float* W1    = (const float*)d_in[4];
    const float* aS1   = (const float*)d_in[5];
    const float* aD1   = (const float*)d_in[6];
    const float* b1    = (const float*)d_in[7];
    const float* W2    = (const float*)d_in[8];
    const float* aS2   = (const float*)d_in[9];
    const float* aD2   = (const float*)d_in[10];
    const float* b2    = (const float*)d_in[11];
    float* out = (float*)d_out;

    const int* srcs = ei;
    const int* dsts = ei + N_EDGES;

    // ---- workspace carve-up (float-sized slots, 256B aligned) ----
    float* ws = (float*)d_ws;
    size_t off = 0;
    auto carve = [&](size_t nelem) {
        size_t o = off;
        off += (nelem + 63) & ~(size_t)63;
        return ws + o;
    };
    float*    h1    = carve((size_t)N_NODES * HIDDEN);   // x @ W1
    float*    als1  = carve((size_t)N_NODES * HEADS);
    float*    ald1  = carve((size_t)N_NODES * HEADS);
    unsigned* menc1 = (unsigned*)carve((size_t)N_NODES * HEADS);
    float*    den1  = carve((size_t)N_NODES * HEADS);
    float*    agg1  = carve((size_t)N_NODES * HIDDEN);   // layer-1 output (post relu)
    float*    h2    = carve((size_t)N_NODES * OUT_DIM);  // agg1 @ W2
    float*    als2  = carve((size_t)N_NODES * HEADS);
    float*    ald2  = carve((size_t)N_NODES * HEADS);
    unsigned* menc2 = (unsigned*)carve((size_t)N_NODES * HEADS);
    float*    den2  = carve((size_t)N_NODES * HEADS);
    float*    agg2  = carve((size_t)N_NODES * OUT_DIM);  // layer-2 output
    float*    psum  = carve((size_t)N_GRAPHS * OUT_DIM);
    float*    pcnt  = carve((size_t)N_GRAPHS);

    const int T = 256;
    auto blks = [&](long long n) { return (unsigned)((n + T - 1) / T); };
    const long long NH   = (long long)N_NODES * HEADS;
    const long long EH   = (long long)E_TOT * HEADS;
    const long long N64  = (long long)N_NODES * HIDDEN;
    const long long N128 = (long long)N_NODES * OUT_DIM;

    // ---- init accumulators (every call: graph-replay safe) ----
    fill_u32<<<blks(NH), T, 0, stream>>>(menc1, 0u, (int)NH);   // encoded -inf
    fill_f32<<<blks(NH), T, 0, stream>>>(den1, 0.0f, (int)NH);
    fill_f32<<<blks(N64), T, 0, stream>>>(agg1, 0.0f, (int)N64);
    fill_u32<<<blks(NH), T, 0, stream>>>(menc2, 0u, (int)NH);
    fill_f32<<<blks(NH), T, 0, stream>>>(den2, 0.0f, (int)NH);
    fill_f32<<<blks(N128), T, 0, stream>>>(agg2, 0.0f, (int)N128);
    fill_f32<<<blks(N_GRAPHS * OUT_DIM), T, 0, stream>>>(psum, 0.0f, N_GRAPHS * OUT_DIM);
    fill_f32<<<blks(N_GRAPHS), T, 0, stream>>>(pcnt, 0.0f, N_GRAPHS);

    // ================= layer 1 =================
    // h1 = x @ W1   (100000x128 @ 128x64): one wave per 16-row strip, NT=4
    gemm_wmma_f32_strip<HIDDEN / 16><<<N_NODES / 16, 32, 0, stream>>>(x, W1, h1, IN_DIM);
    att_logits<<<blks(NH), T, 0, stream>>>(h1, aS1, aD1, als1, ald1, HIDDEN / HEADS);
    edge_max   <<<blks(EH), T, 0, stream>>>(srcs, dsts, als1, ald1, menc1);
    edge_expsum<<<blks(EH), T, 0, stream>>>(srcs, dsts, als1, ald1, menc1, den1);
    edge_agg   <<<blks(EH), T, 0, stream>>>(srcs, dsts, als1, ald1, menc1, den1,
                                            h1, agg1, HIDDEN / HEADS);
    bias_relu<<<blks(N64), T, 0, stream>>>(agg1, b1, HIDDEN - 1, (int)N64);

    // ================= layer 2 =================
    // h2 = agg1 @ W2  (100000x64 @ 64x128): one wave per 16-row strip, NT=8
    gemm_wmma_f32_strip<OUT_DIM / 16><<<N_NODES / 16, 32, 0, stream>>>(agg1, W2, h2, HIDDEN);
    att_logits<<<blks(NH), T, 0, stream>>>(h2, aS2, aD2, als2, ald2, OUT_DIM / HEADS);
    edge_max   <<<blks(EH), T, 0, stream>>>(srcs, dsts, als2, ald2, menc2);
    edge_expsum<<<blks(EH), T, 0, stream>>>(srcs, dsts, als2, ald2, menc2, den2);
    edge_agg   <<<blks(EH), T, 0, stream>>>(srcs, dsts, als2, ald2, menc2, den2,
                                            h2, agg2, OUT_DIM / HEADS);
    bias_relu<<<blks(N128), T, 0, stream>>>(agg2, b2, OUT_DIM - 1, (int)N128);

    // ================= mean pool =================
    pool_acc<<<blks(N128), T, 0, stream>>>(agg2, batch, psum, pcnt);
    pool_div<<<blks(N_GRAPHS * OUT_DIM), T, 0, stream>>>(psum, pcnt, out);
}